// LSTMContextClassifier_33818572489071
// MI455X (gfx1250) — compile-verified
//
#include <hip/hip_runtime.h>

typedef __attribute__((ext_vector_type(2))) float v2f;
typedef __attribute__((ext_vector_type(8))) float v8f;

#define S_  512
#define L_  128
#define D_  256
#define B_  128
#define P_  16
#define H2_ 128
#define F_  256
#define EV_DIM 64
#define FEAT_ (2 * D_ + 2 * EV_DIM)   // 640

// ---------------------------------------------------------------------------
// Kernel 1: gather 16 rows (one attention group), energy GEMM via f32 WMMA,
// softmax over P=16, weighted pool -> feats[:, col_off : col_off+256]
// grid: (B, 2)  block: 256 (8 waves)
// ---------------------------------------------------------------------------
__global__ __launch_bounds__(256) void attn_pool_kernel(
    const int* __restrict__ x,          // [S, L]
    const int* __restrict__ spos,       // [B, P, 2]
    const int* __restrict__ tpos,       // [B, P, 2]
    const float* __restrict__ emb,      // [V, D]
    const float* __restrict__ attw_W,   // [H2, D]
    const float* __restrict__ attw_b,   // [H2]
    const float* __restrict__ attw2_W,  // [H2]
    float* __restrict__ feats)          // [B, FEAT]
{
    __shared__ float lds_h[P_ * D_];        // 16 KB: gathered h tile [16 x 256]
    __shared__ float lds_logit[8][P_];      // per-wave partial logits
    __shared__ float lds_w[P_];             // softmax weights
    __shared__ int   lds_tok[P_];           // gathered token ids

    const int b      = blockIdx.x;
    const int stream = blockIdx.y;          // 0 = spos, 1 = tpos
    const int tid    = threadIdx.x;
    const int wave   = tid >> 5;
    const int lane   = tid & 31;

    const int* pos = (stream == 0) ? spos : tpos;

    // resolve two-level gather: token = x[pos0 * L + pos1]
    if (tid < P_) {
        const int p0 = pos[(b * P_ + tid) * 2 + 0];
        const int p1 = pos[(b * P_ + tid) * 2 + 1];
        lds_tok[tid] = x[p0 * L_ + p1];
    }
    __syncthreads();

    // stage 16 x 256 f32 rows of emb into LDS (coalesced: 256 threads per row)
    #pragma unroll
    for (int p = 0; p < P_; ++p) {
        lds_h[p * D_ + tid] = emb[(long)lds_tok[p] * D_ + tid];
    }
    __syncthreads();

    // ---- energy = tanh(h @ W^T + b), one 16x16 N-tile per wave ----
    // A (16x4 f32): lanes 0-15 -> M=lane, K=kb+0/1 ; lanes 16-31 -> K=kb+2/3
    // B (4x16 f32): N = lane&15, same K split -> float2 from attw_W row
    const int n    = lane & 15;
    const int m    = lane & 15;
    const int koff = (lane >> 4) * 2;
    const int nb   = wave * 16;             // H2 column base for this wave

    const float* arow = lds_h + m * D_ + koff;
    const float* brow = attw_W + (nb + n) * D_ + koff;

    v8f c = {0.f, 0.f, 0.f, 0.f, 0.f, 0.f, 0.f, 0.f};
    #pragma unroll 8
    for (int kb = 0; kb < D_; kb += 4) {
        v2f av = *(const v2f*)(arow + kb);
        v2f bv = *(const v2f*)(brow + kb);
        c = __builtin_amdgcn_wmma_f32_16x16x4_f32(false, av, false, bv,
                                                  (short)0, c, false, false);
    }

    // bias + tanh, then partial logits: part[r] = sum_n energy[m,n]*attw2[n]
    const float bias = attw_b[nb + n];
    const float a2   = attw2_W[nb + n];
    float part[8];
    #pragma unroll
    for (int r = 0; r < 8; ++r) {
        part[r] = tanhf(c[r] + bias) * a2;
    }
    // reduce across the 16 lanes of each half (halves hold different rows)
    #pragma unroll
    for (int mask = 8; mask >= 1; mask >>= 1) {
        #pragma unroll
        for (int r = 0; r < 8; ++r) part[r] += __shfl_xor(part[r], mask, 32);
    }
    if (lane == 0) {
        #pragma unroll
        for (int r = 0; r < 8; ++r) lds_logit[wave][r] = part[r];
    }
    if (lane == 16) {
        #pragma unroll
        for (int r = 0; r < 8; ++r) lds_logit[wave][8 + r] = part[r];
    }
    __syncthreads();

    // softmax over P=16 (tiny: single thread)
    if (tid == 0) {
        float lg[P_];
        float mx = -1e30f;
        for (int p = 0; p < P_; ++p) {
            float s = 0.f;
            for (int w = 0; w < 8; ++w) s += lds_logit[w][p];
            lg[p] = s;
            mx = fmaxf(mx, s);
        }
        float den = 0.f;
        for (int p = 0; p < P_; ++p) { lg[p] = __expf(lg[p] - mx); den += lg[p]; }
        const float inv = 1.f / den;
        for (int p = 0; p < P_; ++p) lds_w[p] = lg[p] * inv;
    }
    __syncthreads();

    // weighted pool over P, one output column per thread
    float acc = 0.f;
    #pragma unroll
    for (int p = 0; p < P_; ++p) acc += lds_w[p] * lds_h[p * D_ + tid];
    feats[b * FEAT_ + stream * D_ + tid] = acc;
}

// ---------------------------------------------------------------------------
// Kernel 2: entity embedding gather -> feats[:, 512:640]
// grid: B  block: 128
// ---------------------------------------------------------------------------
__global__ __launch_bounds__(128) void ent_gather_kernel(
    const int* __restrict__ ents,        // [B, 2]
    const float* __restrict__ ent_emb,   // [E_V, 64]
    float* __restrict__ feats)           // [B, FEAT]
{
    const int b   = blockIdx.x;
    const int tid = threadIdx.x;          // 0..127
    const int which = tid >> 6;           // 0 or 1
    const int e = ents[b * 2 + which];
    feats[b * FEAT_ + 2 * D_ + tid] = ent_emb[(long)e * EV_DIM + (tid & 63)];
}

// ---------------------------------------------------------------------------
// Kernel 3: condensed = tanh(feats @ cond_W^T + b); score = condensed @ lin_W^T + lin_b
// grid: 8 M-tiles (16 batch rows each)  block: 256 (8 waves, 2 N-tiles/wave)
// ---------------------------------------------------------------------------
__global__ __launch_bounds__(256) void condense_kernel(
    const float* __restrict__ feats,     // [B, FEAT]
    const float* __restrict__ cond_W,    // [F, FEAT]
    const float* __restrict__ cond_b,    // [F]
    const float* __restrict__ lin_W,     // [2, F]
    const float* __restrict__ lin_b,     // [2]
    float* __restrict__ out)             // [B, 2]
{
    __shared__ float lds_f[16 * FEAT_];  // 40 KB: feats tile
    __shared__ float lds_c[16 * F_];     // 16 KB: tanh(condensed) tile

    const int mb  = blockIdx.x;
    const int tid = threadIdx.x;
    const int wave = tid >> 5;
    const int lane = tid & 31;

    // stage 16 rows of feats (rows contiguous -> fully linear copy)
    for (int i = tid; i < 16 * FEAT_; i += 256) {
        lds_f[i] = feats[mb * 16 * FEAT_ + i];
    }
    __syncthreads();

    const int n    = lane & 15;
    const int m    = lane & 15;
    const int koff = (lane >> 4) * 2;
    const float* arow = lds_f + m * FEAT_ + koff;

    #pragma unroll
    for (int t = 0; t < 2; ++t) {
        const int ncol = (wave + t * 8) * 16 + n;
        const float* brow = cond_W + ncol * FEAT_ + koff;

        v8f c = {0.f, 0.f, 0.f, 0.f, 0.f, 0.f, 0.f, 0.f};
        #pragma unroll 8
        for (int kb = 0; kb < FEAT_; kb += 4) {
            v2f av = *(const v2f*)(arow + kb);
            v2f bv = *(const v2f*)(brow + kb);
            c = __builtin_amdgcn_wmma_f32_16x16x4_f32(false, av, false, bv,
                                                      (short)0, c, false, false);
        }
        const float bias = cond_b[ncol];
        #pragma unroll
        for (int r = 0; r < 8; ++r) {
            const int row = (lane < 16) ? r : (r + 8);
            lds_c[row * F_ + ncol] = tanhf(c[r] + bias);
        }
    }
    __syncthreads();

    // tiny head: [16 x 256] @ [256 x 2]
    if (tid < 32) {
        const int mrow = tid >> 1;
        const int o    = tid & 1;
        float s = lin_b[o];
        for (int k = 0; k < F_; ++k) s += lds_c[mrow * F_ + k] * lin_W[o * F_ + k];
        out[(mb * 16 + mrow) * 2 + o] = s;
    }
}

// ---------------------------------------------------------------------------
extern "C" void kernel_launch(void* const* d_in, const int* in_sizes, int n_in,
                              void* d_out, int out_size, void* d_ws, size_t ws_size,
                              hipStream_t stream) {
    const int*   x        = (const int*)  d_in[0];
    const int*   ents     = (const int*)  d_in[1];
    const int*   spos     = (const int*)  d_in[2];
    const int*   tpos     = (const int*)  d_in[3];
    const float* emb      = (const float*)d_in[4];
    const float* ent_emb  = (const float*)d_in[5];
    const float* attw_W   = (const float*)d_in[6];
    const float* attw_b   = (const float*)d_in[7];
    const float* attw2_W  = (const float*)d_in[8];
    const float* cond_W   = (const float*)d_in[9];
    const float* cond_b   = (const float*)d_in[10];
    const float* lin_W    = (const float*)d_in[11];
    const float* lin_b    = (const float*)d_in[12];
    float* out   = (float*)d_out;
    float* feats = (float*)d_ws;          // [B, 640] f32 = 320 KB

    attn_pool_kernel<<<dim3(B_, 2), 256, 0, stream>>>(
        x, spos, tpos, emb, attw_W, attw_b, attw2_W, feats);
    ent_gather_kernel<<<dim3(B_), 128, 0, stream>>>(ents, ent_emb, feats);
    condense_kernel<<<dim3(B_ / 16), 256, 0, stream>>>(
        feats, cond_W, cond_b, lin_W, lin_b, out);
}